// GuidedCnn_12575664243367
// MI455X (gfx1250) — compile-verified
//
#include <hip/hip_runtime.h>
#include <hip/hip_bf16.h>

typedef __attribute__((ext_vector_type(2))) float v2f;
typedef __attribute__((ext_vector_type(8))) float v8f;

#define BN   8
#define HH   384
#define WW   384
#define OC   18      // offset-conv output channels
#define KIN  72      // 8 in-channels * 3*3 taps
#define TPR  (WW/16) // 16-pixel tiles per image row = 24

// ---------------------------------------------------------------------------
// im2col gather for the offset conv: x = concat([pf(3), cf(3), mv(2)], C)
// k in [0,72): ci = k/9, tap = k%9, ky = tap/3, kx = tap%3; zero-pad borders.
// ---------------------------------------------------------------------------
__device__ __forceinline__ float im2col_ld(const float* __restrict__ pf,
                                           const float* __restrict__ cf,
                                           const float* __restrict__ mv,
                                           int b, int y, int x, int k) {
    int ci  = k / 9;
    int tap = k - ci * 9;
    int ky  = tap / 3;
    int kx  = tap - ky * 3;
    int yy  = y + ky - 1;
    int xx  = x + kx - 1;
    if ((unsigned)yy >= (unsigned)HH || (unsigned)xx >= (unsigned)WW) return 0.0f;
    const float* base;
    int c, nch;
    if (ci < 3)      { base = pf; c = ci;     nch = 3; }
    else if (ci < 6) { base = cf; c = ci - 3; nch = 3; }
    else             { base = mv; c = ci - 6; nch = 2; }
    return base[(((size_t)b * nch + c) * HH + yy) * (size_t)WW + xx];
}

// ---------------------------------------------------------------------------
// Kernel A: offset = conv3x3(x, offset_w) + offset_b  via V_WMMA_F32_16X16X4
// GEMM view: D[16 pixels][16 ch] += A[16 px][4 k] * B[4 k][16 ch], 18 K-steps,
// two channel-column tiles (ch 0-15, ch 16-17 + zero pad).
// Fragment layouts per CDNA5 ISA 7.12.2:
//   A 16x4 f32 : lane m=lane&15; v0 = K = (lane>=16 ? 2:0), v1 = K+1
//   B  4x16 f32: lane n=lane&15; v0 = row K as above,       v1 = K+1
//   D 16x16 f32: vgpr r; lane<16 -> M=r, N=lane; lane>=16 -> M=r+8, N=lane-16
//
// __launch_bounds__(256, 1): persistent-wave kernel; raise the VGPR budget so
// the 18x2 v2f weight fragments stay register-resident (no scratch spills).
// ---------------------------------------------------------------------------
__global__ void __launch_bounds__(256, 1)
offset_conv_wmma(const float* __restrict__ pf,
                 const float* __restrict__ cf,
                 const float* __restrict__ mv,
                 const float* __restrict__ wgt,   // (18,8,3,3) == 18 x 72
                 const float* __restrict__ bias,  // (18,)
                 float* __restrict__ off,         // (8,18,384,384)
                 int numTiles) {
    const int lane   = threadIdx.x & 31;
    const int m      = lane & 15;          // pixel row of A / channel col of B
    const int hi     = (lane >> 4) & 1;    // second K-pair half of the wave
    const int waveId = blockIdx.x * (blockDim.x >> 5) + (threadIdx.x >> 5);
    const int nWaves = gridDim.x * (blockDim.x >> 5);

    // Preload all B fragments (loop-invariant weights): 18 K-steps x 2 tiles.
    v2f bfrag[18][2];
#pragma unroll
    for (int kk = 0; kk < 18; ++kk) {
        const int k0 = kk * 4 + hi * 2;
#pragma unroll
        for (int t = 0; t < 2; ++t) {
            const int oc = t * 16 + m;
            v2f bv = {0.0f, 0.0f};
            if (oc < OC) {
                bv.x = wgt[oc * KIN + k0];
                bv.y = wgt[oc * KIN + k0 + 1];
            }
            bfrag[kk][t] = bv;
        }
    }
    const float bias0 = bias[m];
    const float bias1 = (m < 2) ? bias[16 + m] : 0.0f;

    for (int tile = waveId; tile < numTiles; tile += nWaves) {
        const int row = tile / TPR;
        const int xb  = (tile - row * TPR) << 4;
        const int b   = row / HH;
        const int y   = row - b * HH;
        const int x   = xb + m;          // this lane's pixel column (A row m)

        v8f acc0 = {};
        v8f acc1 = {};
#pragma unroll
        for (int kk = 0; kk < 18; ++kk) {
            const int k0 = kk * 4 + hi * 2;
            v2f a;
            a.x = im2col_ld(pf, cf, mv, b, y, x, k0);
            a.y = im2col_ld(pf, cf, mv, b, y, x, k0 + 1);
            acc0 = __builtin_amdgcn_wmma_f32_16x16x4_f32(
                false, a, false, bfrag[kk][0], (short)0, acc0, false, false);
            acc1 = __builtin_amdgcn_wmma_f32_16x16x4_f32(
                false, a, false, bfrag[kk][1], (short)0, acc1, false, false);
        }

        // Scatter D back: lane holds N = m; VGPR r holds M = r + hi*8.
#pragma unroll
        for (int r = 0; r < 8; ++r) {
            const int xo = xb + r + hi * 8;
            off[(((size_t)b * OC + m) * HH + y) * (size_t)WW + xo] = acc0[r] + bias0;
            if (m < 2)
                off[(((size_t)b * OC + 16 + m) * HH + y) * (size_t)WW + xo] =
                    acc1[r] + bias1;
        }
    }
}

// ---------------------------------------------------------------------------
// Kernel B: deformable 3x3 conv over pf using the computed offsets.
// Thread = one (b,y,x); fully unrolled taps -> uniform scalar weight loads.
// ---------------------------------------------------------------------------
__device__ __forceinline__ float tap_ld(const float* __restrict__ im, int yy, int xx) {
    if ((unsigned)yy >= (unsigned)HH || (unsigned)xx >= (unsigned)WW) return 0.0f;
    return im[yy * WW + xx];
}

__global__ void deform_conv_kernel(const float* __restrict__ pf,
                                   const float* __restrict__ off,  // (8,18,384,384)
                                   const float* __restrict__ wgt,  // (3,3,3,3)
                                   float* __restrict__ out) {      // (8,3,384,384)
    const int HW  = HH * WW;
    const int idx = blockIdx.x * blockDim.x + threadIdx.x;
    if (idx >= BN * HW) return;
    const int b   = idx / HW;
    const int rem = idx - b * HW;
    const int y   = rem / WW;
    const int x   = rem - y * WW;

    const float* offb = off + (size_t)b * OC * HW + rem;  // channel c at offb[c*HW]
    const float* pfb  = pf  + (size_t)b * 3 * HW;

    float acc0 = 0.0f, acc1 = 0.0f, acc2 = 0.0f;

#pragma unroll
    for (int k = 0; k < 9; ++k) {
        const int ky = k / 3;
        const int kx = k - ky * 3;
        const float py = offb[(size_t)(2 * k)     * HW] + (float)(ky + y - 1);
        const float px = offb[(size_t)(2 * k + 1) * HW] + (float)(kx + x - 1);
        const float fy = floorf(py);
        const float fx = floorf(px);
        const float wy = py - fy;
        const float wx = px - fx;
        const int y0 = (int)fy;
        const int x0 = (int)fx;
        const float w00 = (1.0f - wy) * (1.0f - wx);
        const float w01 = (1.0f - wy) * wx;
        const float w10 = wy * (1.0f - wx);
        const float w11 = wy * wx;
#pragma unroll
        for (int ci = 0; ci < 3; ++ci) {
            const float* im = pfb + (size_t)ci * HW;
            const float v00 = tap_ld(im, y0,     x0);
            const float v01 = tap_ld(im, y0,     x0 + 1);
            const float v10 = tap_ld(im, y0 + 1, x0);
            const float v11 = tap_ld(im, y0 + 1, x0 + 1);
            const float s = v00 * w00 + v01 * w01 + v10 * w10 + v11 * w11;
            acc0 = fmaf(wgt[(0 * 3 + ci) * 9 + k], s, acc0);
            acc1 = fmaf(wgt[(1 * 3 + ci) * 9 + k], s, acc1);
            acc2 = fmaf(wgt[(2 * 3 + ci) * 9 + k], s, acc2);
        }
    }

    out[(size_t)b * 3 * HW + 0 * (size_t)HW + rem] = acc0;
    out[(size_t)b * 3 * HW + 1 * (size_t)HW + rem] = acc1;
    out[(size_t)b * 3 * HW + 2 * (size_t)HW + rem] = acc2;
}

// ---------------------------------------------------------------------------
extern "C" void kernel_launch(void* const* d_in, const int* in_sizes, int n_in,
                              void* d_out, int out_size, void* d_ws, size_t ws_size,
                              hipStream_t stream) {
    const float* pf = (const float*)d_in[0];
    const float* cf = (const float*)d_in[1];
    const float* mv = (const float*)d_in[2];
    const float* ow = (const float*)d_in[3];
    const float* ob = (const float*)d_in[4];
    const float* dw = (const float*)d_in[5];
    float* out = (float*)d_out;
    float* off = (float*)d_ws;   // needs 8*18*384*384*4 B ~ 85 MB (fits in L2)

    const int numTiles = BN * HH * TPR;              // 73728 16-pixel tiles
    // 1024 blocks x 8 waves = 8192 persistent waves, ~9 tiles each
    offset_conv_wmma<<<1024, 256, 0, stream>>>(pf, cf, mv, ow, ob, off, numTiles);

    const int npix = BN * HH * WW;                   // 1.18M pixels
    deform_conv_kernel<<<(npix + 255) / 256, 256, 0, stream>>>(pf, off, dw, out);
}